// Model_60713657697064
// MI455X (gfx1250) — compile-verified
//
#include <hip/hip_runtime.h>
#include <stdint.h>

#define DCH 2048          // channels (last axis)
#define NTHREADS 256      // 8 wave32 waves per block, one row per block
#define EPSV 1e-6f

// CDNA5 async global -> LDS copy, 128 bits per lane, GV addressing (saddr=off).
// Tracked by ASYNCcnt; completion enforced with s_wait_asynccnt.
__device__ __forceinline__ void async_ld_b128(uint32_t lds_off, const void* gptr) {
    asm volatile("global_load_async_to_lds_b128 %0, %1, off"
                 :: "v"(lds_off), "v"(gptr)
                 : "memory");
}

__device__ __forceinline__ void wait_async_zero() {
    asm volatile("s_wait_asynccnt 0x0" ::: "memory");
}

__global__ __launch_bounds__(NTHREADS)
void fused_rms_quant_kernel(const float* __restrict__ x1,
                            const float* __restrict__ x2,
                            const float* __restrict__ gamma,
                            const float* __restrict__ scales,
                            const float* __restrict__ zeropts,
                            float* __restrict__ out) {
    __shared__ float s1[DCH];
    __shared__ float s2[DCH];
    __shared__ float red[NTHREADS / 32];

    const int t    = threadIdx.x;
    const int lane = t & 31;
    const int wave = t >> 5;
    const size_t rowbase = (size_t)blockIdx.x * (size_t)DCH;

    // Two contiguous float4 chunks per thread: channels [c0,c0+4) and [c1,c1+4)
    const int c0 = t * 4;          // covers channels 0..1023
    const int c1 = 1024 + t * 4;   // covers channels 1024..2047

    // ---- Stage both input rows into LDS with async 128-bit copies ----
    async_ld_b128((uint32_t)(uintptr_t)&s1[c0], x1 + rowbase + c0);
    async_ld_b128((uint32_t)(uintptr_t)&s1[c1], x1 + rowbase + c1);
    async_ld_b128((uint32_t)(uintptr_t)&s2[c0], x2 + rowbase + c0);
    async_ld_b128((uint32_t)(uintptr_t)&s2[c1], x2 + rowbase + c1);
    wait_async_zero();
    __syncthreads();

    // ---- residual add (kept in registers) + local sum of squares ----
    const float4 a0 = *(const float4*)&s1[c0];
    const float4 a1 = *(const float4*)&s1[c1];
    const float4 b0 = *(const float4*)&s2[c0];
    const float4 b1 = *(const float4*)&s2[c1];

    float4 xA, xB;
    xA.x = a0.x + b0.x; xA.y = a0.y + b0.y; xA.z = a0.z + b0.z; xA.w = a0.w + b0.w;
    xB.x = a1.x + b1.x; xB.y = a1.y + b1.y; xB.z = a1.z + b1.z; xB.w = a1.w + b1.w;

    float partial = xA.x * xA.x + xA.y * xA.y + xA.z * xA.z + xA.w * xA.w
                  + xB.x * xB.x + xB.y * xB.y + xB.z * xB.z + xB.w * xB.w;

    // ---- wave32 reduction ----
    #pragma unroll
    for (int m = 16; m > 0; m >>= 1)
        partial += __shfl_xor(partial, m, 32);

    if (lane == 0) red[wave] = partial;
    __syncthreads();

    float sum = 0.0f;
    #pragma unroll
    for (int w = 0; w < NTHREADS / 32; ++w)
        sum += red[w];   // conflict-free LDS broadcast

    const float inv_rms = rsqrtf(sum * (1.0f / (float)DCH) + EPSV);

    // ---- normalize, affine, per-channel quantize, clamp, store ----
    const float4 g0  = *(const float4*)(gamma   + c0);
    const float4 g1  = *(const float4*)(gamma   + c1);
    const float4 sc0 = *(const float4*)(scales  + c0);
    const float4 sc1 = *(const float4*)(scales  + c1);
    const float4 z0  = *(const float4*)(zeropts + c0);
    const float4 z1  = *(const float4*)(zeropts + c1);

    float4 q0, q1;
    q0.x = fminf(fmaxf(rintf((xA.x * inv_rms * g0.x) / sc0.x) + z0.x, -128.0f), 127.0f);
    q0.y = fminf(fmaxf(rintf((xA.y * inv_rms * g0.y) / sc0.y) + z0.y, -128.0f), 127.0f);
    q0.z = fminf(fmaxf(rintf((xA.z * inv_rms * g0.z) / sc0.z) + z0.z, -128.0f), 127.0f);
    q0.w = fminf(fmaxf(rintf((xA.w * inv_rms * g0.w) / sc0.w) + z0.w, -128.0f), 127.0f);
    q1.x = fminf(fmaxf(rintf((xB.x * inv_rms * g1.x) / sc1.x) + z1.x, -128.0f), 127.0f);
    q1.y = fminf(fmaxf(rintf((xB.y * inv_rms * g1.y) / sc1.y) + z1.y, -128.0f), 127.0f);
    q1.z = fminf(fmaxf(rintf((xB.z * inv_rms * g1.z) / sc1.z) + z1.z, -128.0f), 127.0f);
    q1.w = fminf(fmaxf(rintf((xB.w * inv_rms * g1.w) / sc1.w) + z1.w, -128.0f), 127.0f);

    *(float4*)(out + rowbase + c0) = q0;
    *(float4*)(out + rowbase + c1) = q1;
}

extern "C" void kernel_launch(void* const* d_in, const int* in_sizes, int n_in,
                              void* d_out, int out_size, void* d_ws, size_t ws_size,
                              hipStream_t stream) {
    const float* x1      = (const float*)d_in[0];
    const float* x2      = (const float*)d_in[1];
    const float* gamma   = (const float*)d_in[2];
    const float* scales1 = (const float*)d_in[3];
    const float* zp1     = (const float*)d_in[4];
    float* out = (float*)d_out;

    const int rows = out_size / DCH;   // B * S = 32768
    fused_rms_quant_kernel<<<rows, NTHREADS, 0, stream>>>(x1, x2, gamma, scales1, zp1, out);
}